// LowPassMSELoss_9904194585239
// MI455X (gfx1250) — compile-verified
//
#include <hip/hip_runtime.h>
#include <hip/hip_bf16.h>

// LowPassMSELoss on MI455X (gfx1250):
//   MSE(lfilter(b,a,output) - lfilter(b,a,target)) = MSE(lfilter(b,a, output-target))
// Block state-space formulation: per 16-sample block,
//   y = L x + G z0 ; z_end = K x + F z0
// L x and K x over 16-block tiles are 16x16x16 f32 GEMMs -> v_wmma_f32_16x16x4_f32 x4.
// Serial state chain handled by hierarchical 6-dim scan.

#define BROWS 16
#define TLEN  262144
#define BLKS_PER_ROW   (TLEN / 16)        // 16384 blocks of 16 samples
#define SB_PER_ROW     (BLKS_PER_ROW / 16) // 1024 super-blocks of 256 samples
#define NTILES         (BROWS * SB_PER_ROW) // 16384 tiles (= super-blocks)
#define NBLOCKS_TOTAL  (BROWS * BLKS_PER_ROW)
#define ORD 6

typedef __attribute__((ext_vector_type(2))) float v2f;
typedef __attribute__((ext_vector_type(8))) float v8f;

// coef buffer layout (floats):
//   [0..255]    L  (16x16 row-major, lower-tri Toeplitz of impulse response)
//   [256..511]  K  (16x16 row-major, rows 0..5 = state-from-input, rest 0)
//   [512..607]  G  (16x6,  G[i*6+s] = output i from unit initial state s)
//   [608..643]  F  (6x6 row-major, 16-step state transition)
//   [644..679]  F16 (= F^16, 256-step state transition)

__device__ __forceinline__ float iir_step(const float* bn, const float* an, float* z, float x) {
    float y = bn[0] * x + z[0];
#pragma unroll
    for (int i = 0; i < ORD - 1; ++i) z[i] = z[i + 1] + bn[i + 1] * x - an[i + 1] * y;
    z[ORD - 1] = bn[ORD] * x - an[ORD] * y;
    return y;
}

__global__ void lp_setup(const float* __restrict__ bc, const float* __restrict__ ac,
                         float* __restrict__ coef) {
    if (threadIdx.x != 0 || blockIdx.x != 0) return;
    float bn[ORD + 1], an[ORD + 1];
    float a0 = ac[0];
    for (int i = 0; i <= ORD; ++i) { bn[i] = bc[i] / a0; an[i] = ac[i] / a0; }
    float* L   = coef;
    float* K   = coef + 256;
    float* G   = coef + 512;
    float* F   = coef + 608;
    float* F16 = coef + 644;
    float z[ORD], h[16];
    // impulse response -> L
    for (int s = 0; s < ORD; ++s) z[s] = 0.f;
    for (int t = 0; t < 16; ++t) h[t] = iir_step(bn, an, z, (t == 0) ? 1.f : 0.f);
    for (int i = 0; i < 16; ++i)
        for (int j = 0; j < 16; ++j) L[i * 16 + j] = (i >= j) ? h[i - j] : 0.f;
    // K: end-state from unit input at position j (padded to 16 rows)
    for (int i = 0; i < 256; ++i) K[i] = 0.f;
    for (int j = 0; j < 16; ++j) {
        for (int s = 0; s < ORD; ++s) z[s] = 0.f;
        for (int t = 0; t < 16; ++t) (void)iir_step(bn, an, z, (t == j) ? 1.f : 0.f);
        for (int s = 0; s < ORD; ++s) K[s * 16 + j] = z[s];
    }
    // G, F from unit initial states
    for (int s = 0; s < ORD; ++s) {
        for (int q = 0; q < ORD; ++q) z[q] = (q == s) ? 1.f : 0.f;
        for (int t = 0; t < 16; ++t) G[t * 6 + s] = iir_step(bn, an, z, 0.f);
        for (int q = 0; q < ORD; ++q) F[q * 6 + s] = z[q];
    }
    // F16 = F^16 via 4 squarings
    float M[36], M2[36];
    for (int i = 0; i < 36; ++i) M[i] = F[i];
    for (int p = 0; p < 4; ++p) {
        for (int r = 0; r < 6; ++r)
            for (int c = 0; c < 6; ++c) {
                float s2 = 0.f;
                for (int k = 0; k < 6; ++k) s2 += M[r * 6 + k] * M[k * 6 + c];
                M2[r * 6 + c] = s2;
            }
        for (int i = 0; i < 36; ++i) M[i] = M2[i];
    }
    for (int i = 0; i < 36; ++i) F16[i] = M[i];
}

// Phase A: per tile (16 blocks = 256 samples), compute forced outputs Yf = L*X and
// forced end-states Zf = K*X with f32 WMMA. One wave per tile.
__global__ void lp_phaseA(const float* __restrict__ out, const float* __restrict__ tgt,
                          const float* __restrict__ coef,
                          float* __restrict__ yf, float* __restrict__ zf) {
    int w     = blockIdx.x * (blockDim.x >> 5) + (threadIdx.x >> 5); // tile id 0..16383
    int lane  = threadIdx.x & 31;
    int laneP = lane & 15;
    int hi    = lane >> 4; // 0 or 1
    long base = (long)(w >> 10) * TLEN + (long)(w & 1023) * 256;
    const float* L = coef;
    const float* K = coef + 256;

    v2f aL[4], aK[4], bx[4];
#pragma unroll
    for (int k = 0; k < 4; ++k) {
        int kk = 4 * k + 2 * hi;
        // A fragment: lane (laneP,hi) holds A[laneP][kk], A[laneP][kk+1]
        aL[k] = v2f{L[laneP * 16 + kk], L[laneP * 16 + kk + 1]};
        aK[k] = v2f{K[laneP * 16 + kk], K[laneP * 16 + kk + 1]};
        // B fragment: B[kk][laneP], B[kk+1][laneP]; X[i][j] = d[base + j*16 + i]
        long idx = base + (long)laneP * 16 + kk;
        float d0 = out[idx]     - tgt[idx];
        float d1 = out[idx + 1] - tgt[idx + 1];
        bx[k] = v2f{d0, d1};
    }

    v8f accY = {};
#pragma unroll
    for (int k = 0; k < 4; ++k)
        accY = __builtin_amdgcn_wmma_f32_16x16x4_f32(false, aL[k], false, bx[k],
                                                     (short)0, accY, false, false);
    v8f accZ = {};
#pragma unroll
    for (int k = 0; k < 4; ++k)
        accZ = __builtin_amdgcn_wmma_f32_16x16x4_f32(false, aK[k], false, bx[k],
                                                     (short)0, accZ, false, false);

    // D layout: vgpr r, lanes 0-15 -> row r, lanes 16-31 -> row r+8; col = laneP.
    long yb = base + (long)laneP * 16 + hi * 8;
#pragma unroll
    for (int r = 0; r < 8; ++r) yf[yb + r] = accY[r];

    if (hi == 0) { // rows 0..5 of Zf = the 6 states for block (w*16 + laneP)
        long zb = ((long)w * 16 + laneP) * 6;
#pragma unroll
        for (int s = 0; s < ORD; ++s) zf[zb + s] = accZ[s];
    }
}

// Phase B1: aggregate forced state across each super-block: z = F z + zf[j], j=0..15
__global__ void lp_phaseB1(const float* __restrict__ zf, const float* __restrict__ coef,
                           float* __restrict__ zsf) {
    int sb = blockIdx.x * blockDim.x + threadIdx.x;
    if (sb >= NTILES) return;
    const float* F = coef + 608;
    float z[ORD] = {0.f, 0.f, 0.f, 0.f, 0.f, 0.f};
    for (int j = 0; j < 16; ++j) {
        const float* zj = zf + ((long)sb * 16 + j) * 6;
        float zn[ORD];
#pragma unroll
        for (int r = 0; r < ORD; ++r) {
            float acc = zj[r];
#pragma unroll
            for (int c = 0; c < ORD; ++c) acc += F[r * 6 + c] * z[c];
            zn[r] = acc;
        }
#pragma unroll
        for (int r = 0; r < ORD; ++r) z[r] = zn[r];
    }
    for (int r = 0; r < ORD; ++r) zsf[(long)sb * 6 + r] = z[r];
}

// Phase B2: per-row scan over 1024 super-blocks: zs[s] = start state of super-block s
__global__ void lp_phaseB2(const float* __restrict__ zsf, const float* __restrict__ coef,
                           float* __restrict__ zs) {
    int row = threadIdx.x;
    if (row >= BROWS || blockIdx.x != 0) return;
    const float* F16 = coef + 644;
    float z[ORD] = {0.f, 0.f, 0.f, 0.f, 0.f, 0.f};
    for (int s = 0; s < SB_PER_ROW; ++s) {
        long sb = (long)row * SB_PER_ROW + s;
        for (int r = 0; r < ORD; ++r) zs[sb * 6 + r] = z[r];
        float zn[ORD];
#pragma unroll
        for (int r = 0; r < ORD; ++r) {
            float acc = zsf[sb * 6 + r];
#pragma unroll
            for (int c = 0; c < ORD; ++c) acc += F16[r * 6 + c] * z[c];
            zn[r] = acc;
        }
#pragma unroll
        for (int r = 0; r < ORD; ++r) z[r] = zn[r];
    }
}

// Phase C: correct y = yf + G z, accumulate sum of squares; deterministic reduction.
__global__ void lp_phaseC(const float* __restrict__ yf, const float* __restrict__ zf,
                          const float* __restrict__ zs, const float* __restrict__ coef,
                          float* __restrict__ partials) {
    int sb = blockIdx.x * blockDim.x + threadIdx.x;
    const float* G = coef + 512;
    const float* F = coef + 608;
    float sum = 0.f;
    if (sb < NTILES) {
        float z[ORD];
        for (int r = 0; r < ORD; ++r) z[r] = zs[(long)sb * 6 + r];
        long base = (long)(sb >> 10) * TLEN + (long)(sb & 1023) * 256;
        for (int j = 0; j < 16; ++j) {
            long b16 = base + (long)j * 16;
#pragma unroll
            for (int i = 0; i < 16; ++i) {
                float y = yf[b16 + i];
#pragma unroll
                for (int s = 0; s < ORD; ++s) y += G[i * 6 + s] * z[s];
                sum += y * y;
            }
            const float* zj = zf + ((long)sb * 16 + j) * 6;
            float zn[ORD];
#pragma unroll
            for (int r = 0; r < ORD; ++r) {
                float acc = zj[r];
#pragma unroll
                for (int c = 0; c < ORD; ++c) acc += F[r * 6 + c] * z[c];
                zn[r] = acc;
            }
#pragma unroll
            for (int r = 0; r < ORD; ++r) z[r] = zn[r];
        }
    }
    __shared__ float sdata[256];
    sdata[threadIdx.x] = sum;
    __syncthreads();
    for (int stride = 128; stride > 0; stride >>= 1) {
        if ((int)threadIdx.x < stride) sdata[threadIdx.x] += sdata[threadIdx.x + stride];
        __syncthreads();
    }
    if (threadIdx.x == 0) partials[blockIdx.x] = sdata[0];
}

__global__ void lp_finalize(const float* __restrict__ partials, float* __restrict__ outp,
                            int nparts) {
    if (threadIdx.x == 0 && blockIdx.x == 0) {
        float s = 0.f;
        for (int i = 0; i < nparts; ++i) s += partials[i];
        outp[0] = s / ((float)BROWS * (float)TLEN);
    }
}

extern "C" void kernel_launch(void* const* d_in, const int* in_sizes, int n_in,
                              void* d_out, int out_size, void* d_ws, size_t ws_size,
                              hipStream_t stream) {
    const float* out_sig = (const float*)d_in[0]; // [16, 262144] f32
    const float* tgt_sig = (const float*)d_in[1]; // [16, 262144] f32
    const float* bcoef   = (const float*)d_in[2]; // [7] f32
    const float* acoef   = (const float*)d_in[3]; // [7] f32
    float* result = (float*)d_out;

    // workspace layout (bytes); total ~23.9 MB
    char* ws = (char*)d_ws;
    const size_t COEF_OFF = 0;
    const size_t YF_OFF   = 4096;
    const size_t ZF_OFF   = YF_OFF  + (size_t)BROWS * TLEN * sizeof(float);        // +16 MB
    const size_t ZSF_OFF  = ZF_OFF  + (size_t)NBLOCKS_TOTAL * 6 * sizeof(float);   // +6.3 MB
    const size_t ZS_OFF   = ZSF_OFF + (size_t)NTILES * 6 * sizeof(float);          // +384 KB
    const size_t PART_OFF = ZS_OFF  + (size_t)NTILES * 6 * sizeof(float);          // +384 KB

    float* coef     = (float*)(ws + COEF_OFF);
    float* yf       = (float*)(ws + YF_OFF);
    float* zf       = (float*)(ws + ZF_OFF);
    float* zsf      = (float*)(ws + ZSF_OFF);
    float* zs       = (float*)(ws + ZS_OFF);
    float* partials = (float*)(ws + PART_OFF);

    lp_setup<<<1, 1, 0, stream>>>(bcoef, acoef, coef);

    // 16384 tiles, one wave32 per tile, 8 waves per 256-thread block
    lp_phaseA<<<NTILES / 8, 256, 0, stream>>>(out_sig, tgt_sig, coef, yf, zf);

    lp_phaseB1<<<NTILES / 256, 256, 0, stream>>>(zf, coef, zsf);
    lp_phaseB2<<<1, BROWS, 0, stream>>>(zsf, coef, zs);

    lp_phaseC<<<NTILES / 256, 256, 0, stream>>>(yf, zf, zs, coef, partials);
    lp_finalize<<<1, 32, 0, stream>>>(partials, result, NTILES / 256);
}